// MultiModalWaveletInterestAttention_24498493456604
// MI455X (gfx1250) — compile-verified
//
#include <hip/hip_runtime.h>

typedef float v2f __attribute__((ext_vector_type(2)));
typedef float v8f __attribute__((ext_vector_type(8)));

#define INV_SQRT2 0.70710678118654752440f
#define BN_EPS 1e-5f
#define SLOPE 0.01f
#define ROWS_PER_BLOCK 128
#define COMB_STRIDE 66   // 64 + 2 pad: A-frag reads stride 66 dwords -> distinct banks
#define W1T_STRIDE 66    // transposed W1 rows, same padding trick
#define HBUF_STRIDE 33   // 32 + 1 pad: per-row reads stride 33 -> distinct banks

// Stage W1 (64x32 row-major) transposed into LDS as W1t[n][k] so B-fragment
// pairs (k, k+1) are contiguous -> ds_load_b64 instead of 2x global_load_b32.
__device__ __forceinline__ void stage_w1t(float* w1t, const float* __restrict__ W1, int t)
{
    for (int idx = t; idx < 64 * 32; idx += 256) {
        const int k = idx >> 5, n = idx & 31;     // coalesced read of W1 row k
        w1t[n * W1T_STRIDE + k] = W1[k * 32 + n];
    }
}

// Compute a 16-row x 32-col tile of h = combined @ W1 for one wave using
// V_WMMA_F32_16X16X4_F32.  comb / w1t are LDS buffers.
// acc0 -> output cols 0..15, acc1 -> cols 16..31.
__device__ __forceinline__ void wmma_h_tile(const float* comb, const float* w1t,
                                            int waveRow, int lane,
                                            v8f& acc0, v8f& acc1)
{
    const int m   = lane & 15;            // A-matrix row within tile
    const int khi = (lane >> 4) << 1;     // lanes 16..31 hold K+2,K+3
    const int n0  = lane & 15;            // B/D column (n-tile 0)
    const int n1  = n0 + 16;              // n-tile 1
    const float* arow = comb + (waveRow + m) * COMB_STRIDE + khi;
    const float* brow0 = w1t + n0 * W1T_STRIDE + khi;
    const float* brow1 = w1t + n1 * W1T_STRIDE + khi;
#pragma unroll
    for (int k0 = 0; k0 < 64; k0 += 4) {
        v2f a;   a.x   = arow[k0];  a.y   = arow[k0 + 1];
        v2f b0;  b0.x  = brow0[k0]; b0.y  = brow0[k0 + 1];
        v2f b1v; b1v.x = brow1[k0]; b1v.y = brow1[k0 + 1];
        acc0 = __builtin_amdgcn_wmma_f32_16x16x4_f32(false, a, false, b0,
                                                     (short)0, acc0, false, false);
        acc1 = __builtin_amdgcn_wmma_f32_16x16x4_f32(false, a, false, b1v,
                                                     (short)0, acc1, false, false);
    }
}

// Pass 1: per-block partial sums of h and h^2 (for BatchNorm over axis 0).
__global__ void __launch_bounds__(256)
k_stats(const float* __restrict__ content, const float* __restrict__ fusion,
        const float* __restrict__ W1, const float* __restrict__ b1,
        float* __restrict__ partial)
{
    __shared__ float comb[ROWS_PER_BLOCK * COMB_STRIDE];
    __shared__ float w1t[32 * W1T_STRIDE];
    __shared__ float lsum[32];
    __shared__ float lsq[32];
    const int t = threadIdx.x;
    if (t < 32) { lsum[t] = 0.0f; lsq[t] = 0.0f; }
    stage_w1t(w1t, W1, t);

    const long long base = (long long)blockIdx.x * ROWS_PER_BLOCK;
    // Stage combined = [cA_c*cA_f | cD_c*cD_f] into LDS (coalesced float2 loads).
    for (int idx = t; idx < ROWS_PER_BLOCK * 32; idx += 256) {
        const int row = idx >> 5, j = idx & 31;
        const float2 c = ((const float2*)(content + (base + row) * 64))[j];
        const float2 f = ((const float2*)(fusion  + (base + row) * 64))[j];
        const float cAc = (c.x + c.y) * INV_SQRT2, cDc = (c.x - c.y) * INV_SQRT2;
        const float cAf = (f.x + f.y) * INV_SQRT2, cDf = (f.x - f.y) * INV_SQRT2;
        comb[row * COMB_STRIDE + j]      = cAc * cAf;   // low_freq
        comb[row * COMB_STRIDE + 32 + j] = cDc * cDf;   // high_freq
    }
    __syncthreads();

    const int lane = t & 31, wave = t >> 5;
    v8f acc0 = {}; v8f acc1 = {};
    wmma_h_tile(comb, w1t, wave * 16, lane, acc0, acc1);

    const int n0 = lane & 15, n1 = n0 + 16;
    const float b1n0 = b1[n0], b1n1 = b1[n1];
    float s0 = 0.f, q0 = 0.f, s1 = 0.f, q1 = 0.f;
#pragma unroll
    for (int r = 0; r < 8; ++r) {
        const float v = acc0[r] + b1n0; s0 += v; q0 += v * v;
        const float w = acc1[r] + b1n1; s1 += w; q1 += w * w;
    }
    atomicAdd(&lsum[n0], s0); atomicAdd(&lsq[n0], q0);
    atomicAdd(&lsum[n1], s1); atomicAdd(&lsq[n1], q1);
    __syncthreads();

    if (t < 32) {
        partial[(size_t)blockIdx.x * 64 + t]      = lsum[t];
        partial[(size_t)blockIdx.x * 64 + 32 + t] = lsq[t];
    }
}

// Pass 2: fold block partials -> mu[32], rsigma[32].  One 1024-thread block:
// 16 slices x 64 features, LDS tree fold, deterministic order.
__global__ void __launch_bounds__(1024)
k_reduce(const float* __restrict__ partial, float* __restrict__ stats,
         int nblocks, float invN)
{
    __shared__ float red[16][64];
    const int f = threadIdx.x & 63;     // 0..31 = sum, 32..63 = sumsq
    const int s = threadIdx.x >> 6;     // slice 0..15
    float acc = 0.f;
    for (int b = s; b < nblocks; b += 16)
        acc += partial[(size_t)b * 64 + f];
    red[s][f] = acc;
    __syncthreads();
    if (threadIdx.x < 64) {
        float tot = 0.f;
#pragma unroll
        for (int i = 0; i < 16; ++i) tot += red[i][f];
        red[0][f] = tot;
    }
    __syncthreads();
    if (threadIdx.x < 32) {
        const float mu  = red[0][threadIdx.x] * invN;
        const float var = red[0][32 + threadIdx.x] * invN - mu * mu;
        stats[threadIdx.x]      = mu;
        stats[32 + threadIdx.x] = rsqrtf(var + BN_EPS);
    }
}

// Pass 3: recompute h via WMMA, BN + LeakyReLU, softmax gate, Haar recon + residual.
__global__ void __launch_bounds__(256)
k_final(const float* __restrict__ content, const float* __restrict__ fusion,
        const float* __restrict__ W1, const float* __restrict__ b1,
        const float* __restrict__ gamma, const float* __restrict__ beta,
        const float* __restrict__ W2, const float* __restrict__ b2,
        const float* __restrict__ stats, float* __restrict__ out)
{
    __shared__ float comb[ROWS_PER_BLOCK * COMB_STRIDE];
    __shared__ float cont[ROWS_PER_BLOCK * 64];
    __shared__ float hbuf[ROWS_PER_BLOCK * HBUF_STRIDE];
    __shared__ float wbuf[ROWS_PER_BLOCK * 2];
    __shared__ float w1t[32 * W1T_STRIDE];
    const int t = threadIdx.x;
    const long long base = (long long)blockIdx.x * ROWS_PER_BLOCK;

    stage_w1t(w1t, W1, t);
    for (int idx = t; idx < ROWS_PER_BLOCK * 32; idx += 256) {
        const int row = idx >> 5, j = idx & 31;
        const float2 c = ((const float2*)(content + (base + row) * 64))[j];
        const float2 f = ((const float2*)(fusion  + (base + row) * 64))[j];
        cont[row * 64 + 2 * j]     = c.x;
        cont[row * 64 + 2 * j + 1] = c.y;
        const float cAc = (c.x + c.y) * INV_SQRT2, cDc = (c.x - c.y) * INV_SQRT2;
        const float cAf = (f.x + f.y) * INV_SQRT2, cDf = (f.x - f.y) * INV_SQRT2;
        comb[row * COMB_STRIDE + j]      = cAc * cAf;
        comb[row * COMB_STRIDE + 32 + j] = cDc * cDf;
    }
    __syncthreads();

    const int lane = t & 31, wave = t >> 5;
    v8f acc0 = {}; v8f acc1 = {};
    wmma_h_tile(comb, w1t, wave * 16, lane, acc0, acc1);

    const int n0 = lane & 15, n1 = n0 + 16;
    const float mu0 = stats[n0],      mu1 = stats[n1];
    const float g0  = gamma[n0] * stats[32 + n0];
    const float g1  = gamma[n1] * stats[32 + n1];
    const float be0 = beta[n0],       be1 = beta[n1];
    const float b1n0 = b1[n0],        b1n1 = b1[n1];
    const int rbase = wave * 16 + ((lane >> 4) << 3);   // D: M = r + 8*lane[4]
#pragma unroll
    for (int r = 0; r < 8; ++r) {
        float v = (acc0[r] + b1n0 - mu0) * g0 + be0;
        v = (v >= 0.f) ? v : SLOPE * v;
        hbuf[(rbase + r) * HBUF_STRIDE + n0] = v;
        float w = (acc1[r] + b1n1 - mu1) * g1 + be1;
        w = (w >= 0.f) ? w : SLOPE * w;
        hbuf[(rbase + r) * HBUF_STRIDE + n1] = w;
    }
    __syncthreads();

    // Per-row 32->2 logits + softmax gate.
    if (t < ROWS_PER_BLOCK) {
        float l0 = b2[0], l1 = b2[1];
        const float* hr = hbuf + t * HBUF_STRIDE;
#pragma unroll
        for (int n = 0; n < 32; ++n) {
            const float hv = hr[n];
            l0 += hv * W2[2 * n];
            l1 += hv * W2[2 * n + 1];
        }
        const float m  = fmaxf(l0, l1);
        const float e0 = __expf(l0 - m), e1 = __expf(l1 - m);
        const float inv = 1.0f / (e0 + e1);
        wbuf[2 * t]     = e0 * inv;
        wbuf[2 * t + 1] = e1 * inv;
    }
    __syncthreads();

    // Haar reconstruction + residual, coalesced float2 stores.
    for (int idx = t; idx < ROWS_PER_BLOCK * 32; idx += 256) {
        const int row = idx >> 5, j = idx & 31;
        const float w0 = wbuf[2 * row], w1 = wbuf[2 * row + 1];
        const float lo = w0 * comb[row * COMB_STRIDE + j];
        const float hi = w1 * comb[row * COMB_STRIDE + 32 + j];
        float2 o;
        o.x = (lo + hi) * INV_SQRT2 + cont[row * 64 + 2 * j];
        o.y = (lo - hi) * INV_SQRT2 + cont[row * 64 + 2 * j + 1];
        ((float2*)(out + (base + row) * 64))[j] = o;
    }
}

extern "C" void kernel_launch(void* const* d_in, const int* in_sizes, int n_in,
                              void* d_out, int out_size, void* d_ws, size_t ws_size,
                              hipStream_t stream)
{
    const float* content = (const float*)d_in[0];
    const float* fusion  = (const float*)d_in[1];
    const float* W1      = (const float*)d_in[2];
    const float* b1      = (const float*)d_in[3];
    const float* gamma   = (const float*)d_in[4];
    const float* beta    = (const float*)d_in[5];
    const float* W2      = (const float*)d_in[6];
    const float* b2      = (const float*)d_in[7];
    float* out = (float*)d_out;

    const int rows    = in_sizes[0] / 64;            // N = 524288
    const int nblocks = rows / ROWS_PER_BLOCK;       // 4096

    float* partial = (float*)d_ws;                   // nblocks*64 floats
    float* stats   = partial + (size_t)nblocks * 64; // 64 floats

    k_stats <<<nblocks, 256, 0, stream>>>(content, fusion, W1, b1, partial);
    k_reduce<<<1, 1024, 0, stream>>>(partial, stats, nblocks, 1.0f / (float)rows);
    k_final <<<nblocks, 256, 0, stream>>>(content, fusion, W1, b1, gamma, beta,
                                          W2, b2, stats, out);
}